// Mixture_PrefixEncoder_15934328669023
// MI455X (gfx1250) — compile-verified
//
#include <hip/hip_runtime.h>
#include <math.h>

// Problem constants (from reference)
#define B_   4
#define S_   2048
#define V_   64
#define D_   1024
#define P_   8
#define H_   42        // D // 24
#define K_   49152     // 2*L*D
#define BV_  256       // B*V
#define HP44 44        // H padded to mult of 4 (WMMA K loop)
#define HP48 48        // H padded to mult of 16 (h storage, 3 N-tiles)
#define EPSF 1e-12f
#define COS_EPSF 1e-8f

typedef float v2f __attribute__((ext_vector_type(2)));
typedef float v8f __attribute__((ext_vector_type(8)));

// Workspace layout (float offsets)
#define WS_V    0          // P*B*S          = 65536
#define WS_COS  65536      // P*B            (padded to 64)
#define WS_GATE 65600      // B*P            (padded to 64)
#define WS_H    65664      // 2048*HP48      = 98304
#define WS_HG   163968     // BV_*HP44       = 11264
// total ~175232 floats ~= 700 KB

// ---------------------------------------------------------------------------
// Kernel 1: v[p,b,s] = dot(ctx[b,s,:], u[p,:]) / max(||ctx[b,s,:]||, EPS)
// where u = e / max(2*|e|, EPS).  One wave per (b,s) row; u staged in LDS.
// ---------------------------------------------------------------------------
__global__ void k1_v(const float* __restrict__ ctx, const float* __restrict__ lora,
                     float* __restrict__ ws) {
  __shared__ float u[P_ * D_];   // 32 KB
  int tid = threadIdx.x;
  for (int i = tid; i < P_ * D_; i += 256) {
    float e = lora[i];
    u[i] = e / fmaxf(2.0f * fabsf(e), EPSF);
  }
  __syncthreads();

  int wave = tid >> 5, lane = tid & 31;
  int row = blockIdx.x * 8 + wave;          // 0 .. B*S-1
  int b = row >> 11, s = row & (S_ - 1);
  const float* cptr = ctx + (size_t)row * D_;

  float ss = 0.f;
  float dp[P_];
#pragma unroll
  for (int p = 0; p < P_; ++p) dp[p] = 0.f;

#pragma unroll 4
  for (int i = 0; i < D_ / 32; ++i) {
    int d = lane + 32 * i;
    float x = cptr[d];
    ss = fmaf(x, x, ss);
#pragma unroll
    for (int p = 0; p < P_; ++p) dp[p] = fmaf(x, u[p * D_ + d], dp[p]);
  }
  // full wave32 reduction
#pragma unroll
  for (int m = 16; m >= 1; m >>= 1) {
    ss += __shfl_xor(ss, m, 32);
#pragma unroll
    for (int p = 0; p < P_; ++p) dp[p] += __shfl_xor(dp[p], m, 32);
  }
  float inv = 1.0f / fmaxf(sqrtf(ss), EPSF);
  if (lane < P_) {
    float val = dp[0];
#pragma unroll
    for (int p = 1; p < P_; ++p) val = (lane == p) ? dp[p] : val;
    ws[WS_V + ((size_t)lane * B_ + b) * S_ + s] = val * inv;
  }
}

// ---------------------------------------------------------------------------
// Kernel 2: per (p,b): softmax over s of n(v), weighted pool a_v, cosine.
// One 256-thread block per (p,b).
// ---------------------------------------------------------------------------
__global__ void k2_av_cos(const float* __restrict__ ctx, const float* __restrict__ lora,
                          float* __restrict__ ws) {
  __shared__ float wsm[S_];    // 8 KB: softmax weights
  __shared__ float red[256];
  int pb = blockIdx.x;         // p*B + b
  int p = pb >> 2, b = pb & 3;
  int tid = threadIdx.x;
  const float* vrow = ws + WS_V + (size_t)pb * S_;
  const float SQS = 45.254833995939045f;   // sqrt(2048)

  float nv[8];
  float lmax = -1e30f;
#pragma unroll
  for (int i = 0; i < 8; ++i) {
    float v = vrow[tid + 256 * i];
    float c = fmaxf(v, 0.f);
    float n = c / fmaxf(SQS * c, EPSF);
    nv[i] = n;
    lmax = fmaxf(lmax, n);
  }
  red[tid] = lmax; __syncthreads();
  for (int o = 128; o >= 1; o >>= 1) {
    if (tid < o) red[tid] = fmaxf(red[tid], red[tid + o]);
    __syncthreads();
  }
  float gmax = red[0]; __syncthreads();

  float lsum = 0.f;
#pragma unroll
  for (int i = 0; i < 8; ++i) {
    float e = expf(nv[i] - gmax);
    wsm[tid + 256 * i] = e;
    lsum += e;
  }
  red[tid] = lsum; __syncthreads();
  for (int o = 128; o >= 1; o >>= 1) {
    if (tid < o) red[tid] += red[tid + o];
    __syncthreads();
  }
  float invs = 1.0f / red[0]; __syncthreads();
#pragma unroll
  for (int i = 0; i < 8; ++i) wsm[tid + 256 * i] *= invs;
  __syncthreads();

  // a_v: thread owns d = 4*tid .. 4*tid+3
  const float4* c4 = (const float4*)(ctx + (size_t)b * S_ * D_);
  float4 acc = make_float4(0.f, 0.f, 0.f, 0.f);
  for (int s = 0; s < S_; ++s) {
    float w = wsm[s];
    float4 x = c4[(size_t)s * (D_ / 4) + tid];
    acc.x = fmaf(w, x.x, acc.x); acc.y = fmaf(w, x.y, acc.y);
    acc.z = fmaf(w, x.z, acc.z); acc.w = fmaf(w, x.w, acc.w);
  }
  float4 e4 = *(const float4*)(lora + (size_t)p * D_ + 4 * tid);
  float pd = e4.x * acc.x + e4.y * acc.y + e4.z * acc.z + e4.w * acc.w;
  float pa = acc.x * acc.x + acc.y * acc.y + acc.z * acc.z + acc.w * acc.w;
  float pe = e4.x * e4.x + e4.y * e4.y + e4.z * e4.z + e4.w * e4.w;

  red[tid] = pd; __syncthreads();
  for (int o = 128; o >= 1; o >>= 1) { if (tid < o) red[tid] += red[tid + o]; __syncthreads(); }
  float dot = red[0]; __syncthreads();
  red[tid] = pa; __syncthreads();
  for (int o = 128; o >= 1; o >>= 1) { if (tid < o) red[tid] += red[tid + o]; __syncthreads(); }
  float a2 = red[0]; __syncthreads();
  red[tid] = pe; __syncthreads();
  for (int o = 128; o >= 1; o >>= 1) { if (tid < o) red[tid] += red[tid + o]; __syncthreads(); }
  float e2 = red[0];

  if (tid == 0)
    ws[WS_COS + pb] = dot / fmaxf(sqrtf(e2) * sqrtf(a2), COS_EPSF);
}

// ---------------------------------------------------------------------------
// Kernel 3: gate[b,p] = softmax(softmax(cos[:,b]))  (double softmax over P)
// ---------------------------------------------------------------------------
__global__ void k3_gate(float* __restrict__ ws) {
  int b = threadIdx.x;
  if (b >= B_) return;
  float c[P_];
  float mx = -1e30f;
#pragma unroll
  for (int p = 0; p < P_; ++p) { c[p] = ws[WS_COS + p * B_ + b]; mx = fmaxf(mx, c[p]); }
  float sum = 0.f;
#pragma unroll
  for (int p = 0; p < P_; ++p) { c[p] = expf(c[p] - mx); sum += c[p]; }
  mx = -1e30f;
#pragma unroll
  for (int p = 0; p < P_; ++p) { c[p] /= sum; mx = fmaxf(mx, c[p]); }
  sum = 0.f;
#pragma unroll
  for (int p = 0; p < P_; ++p) { c[p] = expf(c[p] - mx); sum += c[p]; }
#pragma unroll
  for (int p = 0; p < P_; ++p) ws[WS_GATE + b * P_ + p] = c[p] / sum;
}

// ---------------------------------------------------------------------------
// Kernel 4: h[p,bv,n] = tanh(tables[p, prefix[bv], :] . w1[:, n] + b1[n])
// WMMA f32 16x16x4, one wave per 16x16 tile. M = P*BV = 2048, N = 48 (pad 42).
// Padding guard = clamped UNCONDITIONAL load * loop-invariant 0/1 mask
// (no exec-mask branches in the K loop).
// ---------------------------------------------------------------------------
__global__ void k4_h(const int* __restrict__ prefix, const float* __restrict__ tables,
                     const float* __restrict__ w1, const float* __restrict__ bias1,
                     float* __restrict__ ws) {
  int wid  = (blockIdx.x * blockDim.x + threadIdx.x) >> 5;   // 0..383
  int lane = threadIdx.x & 31;
  int tile_n = wid / 128;                 // 0..2
  int tile_m = wid - tile_n * 128;        // 0..127
  int lane15 = lane & 15;
  int hi = lane >> 4;                     // 0: K=0,1  1: K=2,3 (A/B frag)
  int koff = hi * 2;

  int p  = tile_m >> 4;                                // p constant per M tile
  int bv = ((tile_m & 15) << 4) + lane15;              // 0..255
  const float* arow = tables + ((size_t)p * V_ + prefix[bv]) * D_ + koff;

  int n_g = tile_n * 16 + lane15;
  int nc  = (n_g < H_) ? n_g : (H_ - 1);               // clamped, in-bounds
  float msk = (n_g < H_) ? 1.f : 0.f;                  // loop-invariant mask
  const float* w1c = w1 + nc;

  v8f c = {};
#pragma unroll 4
  for (int k0 = 0; k0 < D_; k0 += 4) {
    float2 af = *(const float2*)(arow + k0);
    v2f a; a[0] = af.x; a[1] = af.y;
    size_t k = (size_t)(k0 + koff);
    v2f bb;
    bb[0] = msk * w1c[k * H_];
    bb[1] = msk * w1c[(k + 1) * H_];
    c = __builtin_amdgcn_wmma_f32_16x16x4_f32(false, a, false, bb, (short)0, c,
                                              false, false);
  }
  float bv1 = msk * bias1[nc];
  float* hout = ws + WS_H;
  int mrow0 = tile_m * 16 + hi * 8;       // C VGPR j -> rows j (lanes<16), j+8
#pragma unroll
  for (int j = 0; j < 8; ++j)
    hout[(size_t)(mrow0 + j) * HP48 + n_g] = tanhf(c[j] + bv1);
}

// ---------------------------------------------------------------------------
// Kernel 4b: hg[bv, j] = sum_p gate[b,p] * h[p*BV+bv, j]   (j padded to 44)
// ---------------------------------------------------------------------------
__global__ void k4b_fold(float* __restrict__ ws) {
  int gid = blockIdx.x * blockDim.x + threadIdx.x;
  if (gid >= BV_ * HP44) return;
  int bv = gid / HP44;
  int j  = gid - bv * HP44;     // cols 42,43 of h are exactly 0 -> hg pad = 0
  int b  = bv >> 6;
  const float* h = ws + WS_H;
  const float* g = ws + WS_GATE + b * P_;
  float acc = 0.f;
#pragma unroll
  for (int p = 0; p < P_; ++p)
    acc = fmaf(g[p], h[(size_t)(p * BV_ + bv) * HP48 + j], acc);
  ws[WS_HG + (size_t)bv * HP44 + j] = acc;
}

// ---------------------------------------------------------------------------
// Kernel 5: out[bv, k] = hg[bv, :] . w2[:, k] + b2[k]
// WMMA f32 16x16x4: M=256 (16 tiles) x N=49152 (3072 tiles), Kdim=44 (11 steps).
// Fully unrolled K loop; pad guard folds to a v_cndmask only in the last step
// (clamped in-bounds load, value select -> no exec branches).
// Bandwidth-bound: streams 8 MB of w2 (L2-resident) and writes 50 MB output.
// ---------------------------------------------------------------------------
__global__ void k5_out(const float* __restrict__ w2, const float* __restrict__ bias2,
                       const float* __restrict__ ws, float* __restrict__ out) {
  int wid  = (blockIdx.x * blockDim.x + threadIdx.x) >> 5;   // 0..49151
  int lane = threadIdx.x & 31;
  int tm = wid & 15;           // fast M => 8 waves of a block share the w2 slice
  int tn = wid >> 4;           // 0..3071
  int lane15 = lane & 15;
  int hi = lane >> 4;
  int koff = hi * 2;

  const float* hg = ws + WS_HG;
  int m   = tm * 16 + lane15;
  int n_g = tn * 16 + lane15;
  const float* arow = hg + (size_t)m * HP44 + koff;
  const float* w2c  = w2 + n_g;

  __builtin_prefetch(w2c, 0, 1);          // global_prefetch_b8 on the B stream

  v8f c = {};
#pragma unroll
  for (int k0 = 0; k0 < HP44; k0 += 4) {
    int k = k0 + koff;
    int kc0 = (k     < H_) ? k     : (H_ - 1);   // clamped, always in-bounds
    int kc1 = (k + 1 < H_) ? k + 1 : (H_ - 1);
    float m0 = (k     < H_) ? 1.f : 0.f;         // constant-folds except k0=40
    float m1 = (k + 1 < H_) ? 1.f : 0.f;
    float2 af = *(const float2*)(arow + k0);
    v2f a; a[0] = af.x; a[1] = af.y;
    v2f bb;
    bb[0] = m0 * w2c[(size_t)kc0 * K_];
    bb[1] = m1 * w2c[(size_t)kc1 * K_];
    c = __builtin_amdgcn_wmma_f32_16x16x4_f32(false, a, false, bb, (short)0, c,
                                              false, false);
  }
  float bv2 = bias2[n_g];
  int mrow0 = tm * 16 + hi * 8;
#pragma unroll
  for (int j = 0; j < 8; ++j)
    out[(size_t)(mrow0 + j) * K_ + n_g] = c[j] + bv2;
}

// ---------------------------------------------------------------------------
extern "C" void kernel_launch(void* const* d_in, const int* in_sizes, int n_in,
                              void* d_out, int out_size, void* d_ws, size_t ws_size,
                              hipStream_t stream) {
  const int*   prefix = (const int*)  d_in[0];   // [B,V]
  const float* ctx    = (const float*)d_in[1];   // [B,S,D]
  const float* tables = (const float*)d_in[2];   // [P,V,D]
  const float* lora   = (const float*)d_in[3];   // [P,D]
  const float* w1     = (const float*)d_in[4];   // [D,H]
  const float* b1     = (const float*)d_in[5];   // [H]
  const float* w2     = (const float*)d_in[6];   // [H,K]
  const float* b2     = (const float*)d_in[7];   // [K]
  float* out = (float*)d_out;                    // [B,V,K]
  float* ws  = (float*)d_ws;

  k1_v     <<<1024, 256, 0, stream>>>(ctx, lora, ws);        // 8192 rows, 1 wave each
  k2_av_cos<<<  32, 256, 0, stream>>>(ctx, lora, ws);        // one block per (p,b)
  k3_gate  <<<   1,  32, 0, stream>>>(ws);
  k4_h     <<<  48, 256, 0, stream>>>(prefix, tables, w1, b1, ws);   // 384 waves
  k4b_fold <<<  44, 256, 0, stream>>>(ws);
  k5_out   <<<6144, 256, 0, stream>>>(w2, b2, ws, out);      // 49152 waves
}